// RetinaFaceFilter_82764019794359
// MI455X (gfx1250) — compile-verified
//
#include <hip/hip_runtime.h>
#include <cstdint>

// RetinaFace post-process for MI455X (gfx1250, wave32).
// No matmul structure -> no WMMA; CDNA5 paths used instead:
//  - global_load_async_to_lds_b128 + s_wait_asynccnt (async staging)
//  - global_prefetch_b8 via __builtin_prefetch
//  - wave32 serial NMS with v_readlane broadcast, state fully in VGPRs
//  - 64KB-LDS single-workgroup bitonic sort (320KB LDS/WGP budget)

#define DEVFN __device__ __forceinline__

constexpr int   B = 8;
constexpr int   P = 85200;
constexpr int   TOP_K = 5000;
constexpr int   KEEP_TOP_K = 750;
constexpr float NMS_THR = 0.4f;
constexpr float CONF_THR = 0.7f;
constexpr float VAR0 = 0.1f, VAR1 = 0.2f;
constexpr int   NBINS = 4096;                 // score histogram bins over (0.7, 1.0]
constexpr int   CAP = 8192;                   // compaction capacity (pow2, sort size)
constexpr int   NCHUNK = (TOP_K + 63) / 64;   // 79 u64 chunks of NMS bitmask
constexpr int   NRW = (TOP_K + 31) / 32;      // 157 u32 words of row-nonempty bitmap
constexpr float NEG_INF = -3.0e38f;

using u32 = unsigned int;
using u64 = unsigned long long;

// Low 32 bits of a generic pointer to LDS == LDS byte offset (aperture scheme).
DEVFN unsigned lds_lo32(const void* p) { return (unsigned)(unsigned long long)p; }

// CDNA5 async copy: 16B global -> LDS per active lane, tracked by ASYNCcnt.
DEVFN void async_copy_b128(unsigned lds_byte_addr, const void* gaddr) {
  asm volatile("global_load_async_to_lds_b128 %0, %1, off"
               :: "v"(lds_byte_addr), "v"((unsigned long long)gaddr)
               : "memory");
}
DEVFN void wait_async0() { asm volatile("s_wait_asynccnt 0" ::: "memory"); }

DEVFN int score_bin(float s) {
  int b = (int)((s - CONF_THR) * ((float)NBINS / (1.0f - CONF_THR)));
  return b < 0 ? 0 : (b > NBINS - 1 ? NBINS - 1 : b);
}

// ---------------- 1) decode boxes, mask scores, histogram ----------------
__global__ void k_decode(const float4* __restrict__ loc, const float2* __restrict__ conf,
                         const float4* __restrict__ priors, const float* __restrict__ scale,
                         float4* __restrict__ boxes, float* __restrict__ scores,
                         u32* __restrict__ hist) {
  int i = blockIdx.x * blockDim.x + threadIdx.x;
  if (i >= B * P) return;
  int b = i / P, p = i - b * P;
  __builtin_prefetch(loc + i + 2048, 0, 3);   // speculative -> global_prefetch_b8
  float4 l = loc[i];
  float4 pr = priors[p];
  float s0 = scale[0], s1 = scale[1], s2 = scale[2], s3 = scale[3];
  float cx = pr.x + l.x * VAR0 * pr.z;
  float cy = pr.y + l.y * VAR0 * pr.w;
  float w  = pr.z * __expf(l.z * VAR1);
  float h  = pr.w * __expf(l.w * VAR1);
  float x1 = cx - 0.5f * w, y1 = cy - 0.5f * h;
  boxes[i] = make_float4(x1 * s0, y1 * s1, (x1 + w) * s2, (y1 + h) * s3);
  float s = conf[i].y;
  scores[i] = (s > CONF_THR) ? s : -1.0f;
  if (s > CONF_THR) atomicAdd(&hist[b * NBINS + score_bin(s)], 1u);
}

// ---------------- 2) per-image threshold bin: largest bin with suffix >= TOP_K ----
__global__ void k_threshold(const u32* __restrict__ hist, int* __restrict__ tbin) {
  int b = blockIdx.x;
  if (threadIdx.x != 0) return;
  u32 acc = 0; int t = 0;
  for (int bin = NBINS - 1; bin >= 0; --bin) {
    acc += hist[b * NBINS + bin];
    if (acc >= (u32)TOP_K) { t = bin; break; }
  }
  tbin[b] = t;  // if total < TOP_K, t==0 -> take everything above CONF_THR
}

// ---------------- 3) compact candidates above threshold bin ----------------
__global__ void k_compact(const float* __restrict__ scores, const int* __restrict__ tbin,
                          u32* __restrict__ cnt, float* __restrict__ ckey,
                          int* __restrict__ cidx) {
  int i = blockIdx.x * blockDim.x + threadIdx.x;
  if (i >= B * P) return;
  int b = i / P, p = i - b * P;
  float s = scores[i];
  if (s > CONF_THR && score_bin(s) >= tbin[b]) {
    u32 pos = atomicAdd(&cnt[b], 1u);
    if (pos < (u32)CAP) { ckey[b * CAP + pos] = s; cidx[b * CAP + pos] = p; }
  }
}

// ---------------- 4) in-LDS bitonic sort (desc by score, asc idx) + gather ----
__global__ __launch_bounds__(1024)
void k_sortgather(const float* __restrict__ ckey, const int* __restrict__ cidx,
                  const u32* __restrict__ cnt, const float4* __restrict__ boxes,
                  float* __restrict__ tops, float4* __restrict__ topb) {
  __shared__ __align__(16) float skey[CAP];   // 32 KB
  __shared__ __align__(16) int   sidx[CAP];   // 32 KB
  int b = blockIdx.x, tid = threadIdx.x;
  // Async-stage both buffers into LDS, 16B per lane per op.
  for (int m = tid; m < CAP / 4; m += blockDim.x) {
    async_copy_b128(lds_lo32(&skey[m * 4]), &ckey[b * CAP + m * 4]);
    async_copy_b128(lds_lo32(&sidx[m * 4]), &cidx[b * CAP + m * 4]);
  }
  wait_async0();
  __syncthreads();
  u32 cv = cnt[b];
  int n = (cv < (u32)CAP) ? (int)cv : CAP;
  for (int k = tid; k < CAP; k += blockDim.x)
    if (k >= n) { skey[k] = NEG_INF; sidx[k] = 0x7FFFFFFF; }
  __syncthreads();
  for (int kk = 2; kk <= CAP; kk <<= 1) {
    for (int j = kk >> 1; j > 0; j >>= 1) {
      for (int i = tid; i < CAP; i += blockDim.x) {
        int ixj = i ^ j;
        if (ixj > i) {
          float a = skey[i], c = skey[ixj];
          int   ai = sidx[i], ci = sidx[ixj];
          bool a_after = (a < c) || (a == c && ai > ci);  // ranks after, descending
          if (((i & kk) == 0) ? a_after : !a_after) {
            skey[i] = c; skey[ixj] = a; sidx[i] = ci; sidx[ixj] = ai;
          }
        }
      }
      __syncthreads();
    }
  }
  for (int k = tid; k < TOP_K; k += blockDim.x) {
    float s = skey[k]; int p = sidx[k];
    tops[b * TOP_K + k] = (s > CONF_THR) ? s : -1.0f;
    topb[b * TOP_K + k] = (p < P) ? boxes[b * P + p] : make_float4(0.f, 0.f, 0.f, 0.f);
  }
}

// ---------------- 5) IoU suppression bitmask + row-nonempty bitmap ----------
__global__ void k_mask(const float4* __restrict__ topb, u64* __restrict__ mask,
                       u32* __restrict__ rowany) {
  int cb = blockIdx.x, rt = blockIdx.y, b = blockIdx.z;
  int tid = threadIdx.x;
  __shared__ __align__(16) float4 sc[64];
  __shared__ float sarea[64];
  int ncols = (TOP_K - cb * 64 < 64) ? (TOP_K - cb * 64) : 64;
  if (tid < ncols)
    async_copy_b128(lds_lo32(&sc[tid]), &topb[b * TOP_K + cb * 64 + tid]);
  wait_async0();
  __syncthreads();
  if (tid < 64) {
    if (tid < ncols) {
      float4 x = sc[tid];
      sarea[tid] = (x.z - x.x + 1.0f) * (x.w - x.y + 1.0f);
    } else {
      sc[tid] = make_float4(0.f, 0.f, -1.0e30f, -1.0e30f);  // zero-IoU sentinel
      sarea[tid] = 0.f;
    }
  }
  __syncthreads();
  int r = rt * blockDim.x + tid;
  if (r >= TOP_K) return;
  float4 rb = topb[b * TOP_K + r];
  float ra = (rb.z - rb.x + 1.0f) * (rb.w - rb.y + 1.0f);
  u64 bits = 0;
#pragma unroll 8
  for (int t = 0; t < 64; ++t) {
    float4 cbx = sc[t];
    float xx1 = fmaxf(rb.x, cbx.x), yy1 = fmaxf(rb.y, cbx.y);
    float xx2 = fminf(rb.z, cbx.z), yy2 = fminf(rb.w, cbx.w);
    float w = fmaxf(0.f, xx2 - xx1 + 1.0f);
    float h = fmaxf(0.f, yy2 - yy1 + 1.0f);
    float inter = w * h;
    float iou = inter / (ra + sarea[t] - inter);
    if (iou > NMS_THR && (cb * 64 + t) != r) bits |= (1ull << t);
  }
  mask[((size_t)(b * TOP_K + r)) * NCHUNK + cb] = bits;
  if (bits) atomicOr(&rowany[b * NRW + (r >> 5)], 1u << (r & 31));
}

// ---------------- 6) serial greedy NMS: one wave32 per image, state in VGPRs ---
__global__ void k_reduce(const float* __restrict__ tops, const u64* __restrict__ mask,
                         const u32* __restrict__ rowany, u64* __restrict__ keep) {
  int b = blockIdx.x;
  int lane = threadIdx.x;                 // 32 threads: chunk c owned by lane c&31, slot c>>5
  u64 remv0 = 0, remv1 = 0, remv2 = 0;    // suppression bitmap (79 chunks striped)
  u64 k0 = 0, k1 = 0, k2 = 0;             // keep bitmap
  u64 v0 = 0, v1 = 0, v2 = 0;             // validity (score > CONF_THR) bitmap
  const float* ts = tops + b * TOP_K;
  for (int s = 0; s < 3; ++s) {
    int c = lane + 32 * s;
    if (c < NCHUNK) {
      u64 v = 0;
      int base = c * 64;
      for (int t = 0; t < 64; ++t) {
        int p = base + t;
        if (p < TOP_K && ts[p] > CONF_THR) v |= (1ull << t);
      }
      if (s == 0) v0 = v; else if (s == 1) v1 = v; else v2 = v;
    }
  }
  u32 a0 = 0, a1 = 0, a2 = 0, a3 = 0, a4 = 0;   // row-nonempty bitmap (157 words striped)
  for (int s = 0; s < 5; ++s) {
    int w = lane + 32 * s;
    u32 v = (w < NRW) ? rowany[b * NRW + w] : 0u;
    if (s == 0) a0 = v; else if (s == 1) a1 = v; else if (s == 2) a2 = v;
    else if (s == 3) a3 = v; else a4 = v;
  }
  const u64* mbase = mask + (size_t)b * TOP_K * NCHUNK;
  for (int i = 0; i < TOP_K; ++i) {
    int c = i >> 6, bit = i & 63;
    int slot = c >> 5, owner = c & 31;                      // wave-uniform slot
    u64 rsel = (slot == 0) ? remv0 : (slot == 1) ? remv1 : remv2;
    u64 vsel = (slot == 0) ? v0 : (slot == 1) ? v1 : v2;
    unsigned flags = (unsigned)((rsel >> bit) & 1ull) |
                     ((unsigned)((vsel >> bit) & 1ull) << 1);
    unsigned f = (unsigned)__builtin_amdgcn_readlane((int)flags, owner);
    bool keepi = (f & 2u) && !(f & 1u);
    if (keepi) {
      if (lane == owner) {
        u64 m = 1ull << bit;
        if (slot == 0) k0 |= m; else if (slot == 1) k1 |= m; else k2 |= m;
      }
      int wdx = i >> 5;
      int aslot = wdx >> 5, aowner = wdx & 31;
      u32 asel = (aslot == 0) ? a0 : (aslot == 1) ? a1 : (aslot == 2) ? a2
               : (aslot == 3) ? a3 : a4;
      unsigned any = (unsigned)__builtin_amdgcn_readlane(
                         (int)((asel >> (wdx & 31)) & 1u), aowner);
      if (any) {                                            // rare: OR suppression row
        const u64* row = mbase + (size_t)i * NCHUNK;
        remv0 |= row[lane];
        remv1 |= row[lane + 32];
        if (lane < NCHUNK - 64) remv2 |= row[lane + 64];
      }
    }
  }
  keep[b * NCHUNK + lane]      = k0;
  keep[b * NCHUNK + lane + 32] = k1;
  if (lane < NCHUNK - 64) keep[b * NCHUNK + lane + 64] = k2;
}

// ---------------- 7) rank kept boxes, emit first 750 rows ----------------
__global__ void k_finalize(const u64* __restrict__ keep, const float* __restrict__ tops,
                           const float4* __restrict__ topb, float* __restrict__ out) {
  int b = blockIdx.x, tid = threadIdx.x;
  __shared__ u32 cnts[NCHUNK];
  __shared__ u32 pref[NCHUNK];
  for (int k = tid; k < KEEP_TOP_K * 5; k += blockDim.x)
    out[(size_t)b * KEEP_TOP_K * 5 + k] = 0.f;
  if (tid < NCHUNK) cnts[tid] = __popcll(keep[b * NCHUNK + tid]);
  __syncthreads();
  if (tid == 0) {
    u32 acc = 0;
    for (int c = 0; c < NCHUNK; ++c) { pref[c] = acc; acc += cnts[c]; }
  }
  __syncthreads();
  if (tid < NCHUNK) {
    u64 bits = keep[b * NCHUNK + tid];
    u32 r = pref[tid];
    int base = tid * 64;
    while (bits) {
      int t = __ffsll((unsigned long long)bits) - 1;
      bits &= bits - 1;
      if (r < (u32)KEEP_TOP_K) {
        int i = base + t;
        float4 bx = topb[b * TOP_K + i];
        float* o = out + (size_t)b * KEEP_TOP_K * 5 + (size_t)r * 5;
        o[0] = bx.x; o[1] = bx.y; o[2] = bx.z; o[3] = bx.w;
        o[4] = tops[b * TOP_K + i];
      }
      r++;
    }
  }
}

extern "C" void kernel_launch(void* const* d_in, const int* in_sizes, int n_in,
                              void* d_out, int out_size, void* d_ws, size_t ws_size,
                              hipStream_t stream) {
  (void)in_sizes; (void)n_in; (void)out_size; (void)ws_size;
  const float4* loc    = (const float4*)d_in[0];
  const float2* conf   = (const float2*)d_in[1];
  const float4* priors = (const float4*)d_in[2];
  const float*  scale  = (const float*)d_in[3];

  char* ws = (char*)d_ws;
  size_t off = 0;
  auto take = [&](size_t bytes) -> void* {
    void* p = ws + off;
    off += (bytes + 255) & ~(size_t)255;
    return p;
  };
  float4* boxes  = (float4*)take(sizeof(float4) * (size_t)B * P);        // 10.9 MB
  float*  scores = (float*) take(sizeof(float)  * (size_t)B * P);        //  2.7 MB
  u32*    hist   = (u32*)   take(sizeof(u32) * B * NBINS);               //  128 KB
  int*    tbin   = (int*)   take(sizeof(int) * B);
  u32*    cnt    = (u32*)   take(sizeof(u32) * B);
  float*  ckey   = (float*) take(sizeof(float) * B * CAP);               //  256 KB
  int*    cidx   = (int*)   take(sizeof(int)   * B * CAP);               //  256 KB
  float*  tops   = (float*) take(sizeof(float)  * B * TOP_K);            //  160 KB
  float4* topb   = (float4*)take(sizeof(float4) * B * TOP_K);            //  640 KB
  u64*    mask   = (u64*)   take(sizeof(u64) * (size_t)B * TOP_K * NCHUNK); // 25.3 MB
  u32*    rowany = (u32*)   take(sizeof(u32) * B * NRW);
  u64*    keepbm = (u64*)   take(sizeof(u64) * B * NCHUNK);

  hipMemsetAsync(hist,   0, sizeof(u32) * B * NBINS, stream);
  hipMemsetAsync(cnt,    0, sizeof(u32) * B, stream);
  hipMemsetAsync(rowany, 0, sizeof(u32) * B * NRW, stream);

  const int nBP = B * P;
  k_decode   <<<(nBP + 255) / 256, 256, 0, stream>>>(loc, conf, priors, scale,
                                                     boxes, scores, hist);
  k_threshold<<<B, 32, 0, stream>>>(hist, tbin);
  k_compact  <<<(nBP + 255) / 256, 256, 0, stream>>>(scores, tbin, cnt, ckey, cidx);
  k_sortgather<<<B, 1024, 0, stream>>>(ckey, cidx, cnt, boxes, tops, topb);
  k_mask     <<<dim3(NCHUNK, (TOP_K + 255) / 256, B), 256, 0, stream>>>(topb, mask, rowany);
  k_reduce   <<<B, 32, 0, stream>>>(tops, mask, rowany, keepbm);
  k_finalize <<<B, 256, 0, stream>>>(keepbm, tops, topb, (float*)d_out);
}